// Blur_87548613362427
// MI455X (gfx1250) — compile-verified
//
#include <hip/hip_runtime.h>

// Blur(kernel_size=4, up=2, down=1) == upfirdn2d, input (16,128,128,128) f32,
// output (16,128,255,255) f32. Each output pixel = weighted 2x2 input patch,
// weights chosen by output-coordinate parity from the 4x4 binomial kernel.
//
// Memory-bound: ~667 MB traffic, ~1 GFLOP -> target is HBM bandwidth (23.3 TB/s,
// ~29 us). Optimization strategy for MI455X:
//   * stage 65x9 input tiles into LDS with gfx1250 async global->LDS DMA
//     (GLOBAL_LOAD_ASYNC_TO_LDS_B32 / ASYNCcnt / s_wait_asynccnt),
//   * 4 FMAs per output from LDS, 2x2 quad per thread,
//   * nontemporal output stores so the 134 MB input stays resident in the
//     192 MB L2 while 533 MB of output streams through.

namespace {
constexpr int IN_H = 128, IN_W = 128;
constexpr int OUT_H = 255, OUT_W = 255;
constexpr int OW_TILE = 128, OH_TILE = 16;
constexpr int IW_TILE = OW_TILE / 2 + 1;  // 65 input cols per tile
constexpr int IH_TILE = OH_TILE / 2 + 1;  // 9 input rows per tile
constexpr int LDSS = IW_TILE + 1;         // 66: padded stride, conflict-free
}  // namespace

typedef __attribute__((address_space(3))) float lds_f32;
typedef __attribute__((address_space(3))) int lds_i32;
typedef __attribute__((address_space(1))) int g_i32;

__global__ __launch_bounds__(256) void Blur_87548613362427_kernel(
    const float* __restrict__ in, const float* __restrict__ kern,
    float* __restrict__ out) {
  __shared__ float tile[IH_TILE * LDSS];

  const int tid = threadIdx.x;
  const int bx = blockIdx.x;  // 0..1   (x tiles)
  const int by = blockIdx.y;  // 0..15  (y tiles)
  const int nc = blockIdx.z;  // 0..N*C-1

  const int c0 = bx * (OW_TILE / 2);
  const int r0 = by * (OH_TILE / 2);

  const float* gin = in + (size_t)nc * (IN_H * IN_W);

  // ---- stage input tile into LDS (zero-pad out-of-range slots) ----
  for (int i = tid; i < IH_TILE * IW_TILE; i += 256) {
    const int rr = i / IW_TILE;
    const int cc = i - rr * IW_TILE;
    const int r = r0 + rr;
    const int c = c0 + cc;
    lds_f32* dst = (lds_f32*)&tile[rr * LDSS + cc];
    if ((r < IN_H) && (c < IN_W)) {
#if __has_builtin(__builtin_amdgcn_global_load_async_to_lds_b32)
      __builtin_amdgcn_global_load_async_to_lds_b32(
          (g_i32*)(gin + r * IN_W + c), (lds_i32*)dst, 0, 0);
#else
      *dst = gin[r * IN_W + c];
#endif
    } else {
      *dst = 0.0f;  // disjoint LDS addresses from the async path: no race
    }
  }
#if __has_builtin(__builtin_amdgcn_s_wait_asynccnt)
  __builtin_amdgcn_s_wait_asynccnt(0);
#else
  asm volatile("s_wait_asynccnt 0x0" ::: "memory");
#endif
  __syncthreads();

  // ---- kernel weights K[sy][sx] (uniform broadcast loads) ----
  const float4* k4 = (const float4*)kern;
  const float4 k0 = k4[0], k1 = k4[1], k2 = k4[2], k3 = k4[3];

  float* gout = out + (size_t)nc * (OUT_H * OUT_W);

  const int qx = tid & 63;
  const int qy0 = tid >> 6;  // 0..3; handles quad rows qy0 and qy0+4

#pragma unroll
  for (int qq = 0; qq < 2; ++qq) {
    const int qy = qy0 + qq * 4;

    const float a = tile[qy * LDSS + qx];
    const float b = tile[qy * LDSS + qx + 1];
    const float c = tile[(qy + 1) * LDSS + qx];
    const float d = tile[(qy + 1) * LDSS + qx + 1];

    // output quad at (oy, ox); parities select kernel taps:
    //  even coord -> taps {1,3}, odd coord -> taps {0,2}
    const float o00 = k1.y * a + k1.w * b + k3.y * c + k3.w * d;  // (even,even)
    const float o01 = k1.x * a + k1.z * b + k3.x * c + k3.z * d;  // (even,odd)
    const float o10 = k0.y * a + k0.w * b + k2.y * c + k2.w * d;  // (odd,even)
    const float o11 = k0.x * a + k0.z * b + k2.x * c + k2.z * d;  // (odd,odd)

    const int oy = by * OH_TILE + 2 * qy;      // <= 254 always
    const int ox = bx * OW_TILE + 2 * qx;      // <= 254 always
    float* row0 = gout + (size_t)oy * OUT_W + ox;

    __builtin_nontemporal_store(o00, row0);
    if (ox + 1 < OUT_W) __builtin_nontemporal_store(o01, row0 + 1);
    if (oy + 1 < OUT_H) {
      float* row1 = row0 + OUT_W;
      __builtin_nontemporal_store(o10, row1);
      if (ox + 1 < OUT_W) __builtin_nontemporal_store(o11, row1 + 1);
    }
  }
}

extern "C" void kernel_launch(void* const* d_in, const int* in_sizes, int n_in,
                              void* d_out, int out_size, void* d_ws, size_t ws_size,
                              hipStream_t stream) {
  const float* imgs = (const float*)d_in[0];
  const float* kern = (const float*)d_in[1];
  float* out = (float*)d_out;

  const int nc = in_sizes[0] / (IN_H * IN_W);  // N*C = 2048
  dim3 grid(2, 16, nc);
  Blur_87548613362427_kernel<<<grid, 256, 0, stream>>>(imgs, kern, out);
}